// Attention_19782619365760
// MI455X (gfx1250) — compile-verified
//
#include <hip/hip_runtime.h>
#include <math.h>

typedef __attribute__((ext_vector_type(2))) float v2f;
typedef __attribute__((ext_vector_type(8))) float v8f;
typedef __attribute__((ext_vector_type(4))) int   v4i;

#define TM 128
#define TN 64
#define TK 16
#define ASS 20   // A LDS row stride (floats): 16B-aligned rows, conflict-free frags
#define BSS 20   // B[n][k] stride (b_trans path)
#define BSN 68   // B[k][n] stride (no-trans path)
#define BBUF 1280  // max(64*BSS, 16*BSN) floats

#if defined(__has_builtin)
#if __has_builtin(__builtin_amdgcn_global_load_async_to_lds_b128) && \
    __has_builtin(__builtin_amdgcn_s_wait_asynccnt)
#define USE_ASYNC 1
#endif
#endif

// async builtin signature (from hipcc diagnostics): param0 = v4i addrspace(1)*,
// param1 = v4i addrspace(3)*, then (imm offset, imm cpol).
#define GP(p) ((__attribute__((address_space(1))) v4i*)(p))
#define LP(p) ((__attribute__((address_space(3))) v4i*)(p))

__device__ __forceinline__ int imin(int a, int b) { return a < b ? a : b; }

// ---------------------------------------------------------------------------
// Build qkv bias vector: [q_bias(768) | zeros(768) | v_bias(768)]
// ---------------------------------------------------------------------------
__global__ void build_qkv_bias(const float* __restrict__ q_bias,
                               const float* __restrict__ v_bias,
                               float* __restrict__ out, int C) {
    int i = blockIdx.x * blockDim.x + threadIdx.x;
    if (i >= 3 * C) return;
    float v = 0.f;
    if (i < C) v = q_bias[i];
    else if (i >= 2 * C) v = v_bias[i - 2 * C];
    out[i] = v;
}

// ---------------------------------------------------------------------------
// Batched GEMM on V_WMMA_F32_16X16X4_F32, double-buffered LDS staging with
// b128 moves (async global->LDS DMA when the toolchain exposes it).
//   BT=1: B stored [Ncols,K] (weights);  BT=0: B stored [K,Ncols].
//   Slice offset = (s/slice_div)*outer + (s%slice_div)*inner.
//   mask row (if present) = mask + (s/slice_div)*mask_ld; mask[n]==0 -> neg.
// Block: 256 threads = 8 waves; block tile 128x64; wave tile 32x32 (2x2 WMMA).
// ---------------------------------------------------------------------------
template <int BT>
__global__ __launch_bounds__(256)
void gemm_wmma_f32(const float* __restrict__ A, long long a_outer, long long a_inner, int lda,
                   const float* __restrict__ B, long long b_outer, long long b_inner, int ldb,
                   float* __restrict__ C, long long c_outer, long long c_inner, int ldc,
                   const float* __restrict__ bias,
                   const int* __restrict__ mask, int mask_ld,
                   float scale, float neg_val,
                   int M, int Ncols, int K, int slice_div) {
    __shared__ float As[2][TM * ASS];
    __shared__ float Bs[2][BBUF];

    const int s  = blockIdx.z;
    const int so = s / slice_div;
    const int si = s - so * slice_div;
    A += so * a_outer + si * a_inner;
    B += so * b_outer + si * b_inner;
    C += so * c_outer + si * c_inner;
    const int* mrow = mask ? (mask + (long long)so * mask_ld) : (const int*)0;

    const int gm0  = blockIdx.y * TM;
    const int gn0  = blockIdx.x * TN;
    const int tid  = threadIdx.x;
    const int lane = tid & 31;
    const int wave = tid >> 5;
    const int wm0  = (wave >> 1) * 32;   // 4 waves along M
    const int wn0  = (wave & 1) * 32;    // 2 waves along N

    // --- per-thread staging geometry (all b128 chunks) ---------------------
    // A tile: 128 rows x 4 chunks = 512 b128, 2 per thread.
    const int ar0 = tid >> 2,             akq0 = tid & 3;
    const int ar1 = (tid + 256) >> 2,     akq1 = (tid + 256) & 3;
    // B tile, BT=1: 64 rows x 4 chunks = 256 b128, 1 per thread.
    const int br  = tid >> 2,             bkq  = tid & 3;
    const int brow = imin(gn0 + br, Ncols - 1);          // clamp (harmless dup)
    // B tile, BT=0: 16 k-rows x 16 chunks = 256 b128, 1 per thread.
    const int bk  = tid >> 4,             bnq  = tid & 15;
    const int bn  = imin(gn0 + bnq * 4, Ncols - 4);      // clamp (harmless dup)

    const int T = K / TK;

    v8f acc00 = {}, acc01 = {}, acc10 = {}, acc11 = {};
    const int lm   = lane & 15;
    const int koff = (lane >> 4) << 1;   // lanes 16-31 hold k+2,k+3

#if USE_ASYNC
    // ---- async global->LDS DMA, ASYNCcnt-tracked, double buffered ---------
    auto issue = [&](int kt, int buf) {
        __builtin_amdgcn_global_load_async_to_lds_b128(
            GP(A + (long long)(gm0 + ar0) * lda + kt + akq0 * 4),
            LP(&As[buf][ar0 * ASS + akq0 * 4]), 0, 0);
        __builtin_amdgcn_global_load_async_to_lds_b128(
            GP(A + (long long)(gm0 + ar1) * lda + kt + akq1 * 4),
            LP(&As[buf][ar1 * ASS + akq1 * 4]), 0, 0);
        if (BT) {
            __builtin_amdgcn_global_load_async_to_lds_b128(
                GP(B + (long long)brow * ldb + kt + bkq * 4),
                LP(&Bs[buf][br * BSS + bkq * 4]), 0, 0);
        } else {
            __builtin_amdgcn_global_load_async_to_lds_b128(
                GP(B + (long long)(kt + bk) * ldb + bn),
                LP(&Bs[buf][bk * BSN + bnq * 4]), 0, 0);
        }
    };
    issue(0, 0);
#else
    // ---- fallback: b128 register prefetch, double buffered ----------------
    float4 aPre0, aPre1, bPre;
    auto fetch = [&](int kt) {
        aPre0 = *(const float4*)(A + (long long)(gm0 + ar0) * lda + kt + akq0 * 4);
        aPre1 = *(const float4*)(A + (long long)(gm0 + ar1) * lda + kt + akq1 * 4);
        if (BT) bPre = *(const float4*)(B + (long long)brow * ldb + kt + bkq * 4);
        else    bPre = *(const float4*)(B + (long long)(kt + bk) * ldb + bn);
    };
    auto stash = [&](int buf) {
        *(float4*)&As[buf][ar0 * ASS + akq0 * 4] = aPre0;
        *(float4*)&As[buf][ar1 * ASS + akq1 * 4] = aPre1;
        if (BT) *(float4*)&Bs[buf][br * BSS + bkq * 4] = bPre;
        else    *(float4*)&Bs[buf][bk * BSN + bnq * 4] = bPre;
    };
    fetch(0);
#endif

    for (int t = 0; t < T; ++t) {
        const int buf = t & 1;
#if USE_ASYNC
        if (t + 1 < T) {
            issue((t + 1) * TK, buf ^ 1);
            __builtin_amdgcn_s_wait_asynccnt(3);   // tile t complete, t+1 in flight
        } else {
            __builtin_amdgcn_s_wait_asynccnt(0);
        }
        __syncthreads();
#else
        stash(buf);
        __syncthreads();
        if (t + 1 < T) fetch((t + 1) * TK);
#endif
        const float* as = As[buf];
        const float* bs = Bs[buf];
        #pragma unroll
        for (int kk = 0; kk < TK; kk += 4) {
            const int ko = kk + koff;
            v2f a0, a1, b0, b1;
            a0.x = as[(wm0 + lm     ) * ASS + ko];
            a0.y = as[(wm0 + lm     ) * ASS + ko + 1];
            a1.x = as[(wm0 + 16 + lm) * ASS + ko];
            a1.y = as[(wm0 + 16 + lm) * ASS + ko + 1];
            if (BT) {
                b0.x = bs[(wn0 + lm     ) * BSS + ko];
                b0.y = bs[(wn0 + lm     ) * BSS + ko + 1];
                b1.x = bs[(wn0 + 16 + lm) * BSS + ko];
                b1.y = bs[(wn0 + 16 + lm) * BSS + ko + 1];
            } else {
                b0.x = bs[(ko    ) * BSN + (wn0 + lm)];
                b0.y = bs[(ko + 1) * BSN + (wn0 + lm)];
                b1.x = bs[(ko    ) * BSN + (wn0 + 16 + lm)];
                b1.y = bs[(ko + 1) * BSN + (wn0 + 16 + lm)];
            }
            acc00 = __builtin_amdgcn_wmma_f32_16x16x4_f32(false, a0, false, b0, (short)0, acc00, false, false);
            acc01 = __builtin_amdgcn_wmma_f32_16x16x4_f32(false, a0, false, b1, (short)0, acc01, false, false);
            acc10 = __builtin_amdgcn_wmma_f32_16x16x4_f32(false, a1, false, b0, (short)0, acc10, false, false);
            acc11 = __builtin_amdgcn_wmma_f32_16x16x4_f32(false, a1, false, b1, (short)0, acc11, false, false);
        }
        __syncthreads();
    }

    // Epilogue. C/D layout: VGPR v holds (M = v + 8*(lane>=16), N = lane&15).
    const int ln = lane & 15;
    const int mo = (lane >> 4) * 8;
    v8f accs[2][2] = {{acc00, acc01}, {acc10, acc11}};
    #pragma unroll
    for (int ti = 0; ti < 2; ++ti) {
        #pragma unroll
        for (int tj = 0; tj < 2; ++tj) {
            int ng = gn0 + wn0 + tj * 16 + ln;
            if (ng >= Ncols) continue;
            float bval   = bias ? bias[ng] : 0.f;
            bool  masked = mrow ? (mrow[ng] == 0) : false;
            #pragma unroll
            for (int v = 0; v < 8; ++v) {
                int mg = gm0 + wm0 + ti * 16 + mo + v;
                if (mg >= M) continue;
                float val = accs[ti][tj][v] * scale + bval;
                if (masked) val = neg_val;
                C[(long long)mg * ldc + ng] = val;
            }
        }
    }
}

// ---------------------------------------------------------------------------
// In-place row softmax; one 256-thread block per row of length rowlen (2048).
// ---------------------------------------------------------------------------
__global__ __launch_bounds__(256)
void softmax_rows(float* __restrict__ attn, int rowlen) {
    float* p = attn + (long long)blockIdx.x * rowlen;
    const int tid = threadIdx.x;
    const int iters = rowlen >> 8;   // 8 for 2048
    float v[8];
    float mx = -INFINITY;
    for (int j = 0; j < iters; ++j) {
        v[j] = p[tid + j * 256];
        mx = fmaxf(mx, v[j]);
    }
    __shared__ float red[256];
    red[tid] = mx;
    __syncthreads();
    for (int s = 128; s > 0; s >>= 1) {
        if (tid < s) red[tid] = fmaxf(red[tid], red[tid + s]);
        __syncthreads();
    }
    mx = red[0];
    __syncthreads();
    float sum = 0.f;
    for (int j = 0; j < iters; ++j) {
        v[j] = __expf(v[j] - mx);
        sum += v[j];
    }
    red[tid] = sum;
    __syncthreads();
    for (int s = 128; s > 0; s >>= 1) {
        if (tid < s) red[tid] += red[tid + s];
        __syncthreads();
    }
    float inv = 1.0f / red[0];
    for (int j = 0; j < iters; ++j) p[tid + j * 256] = v[j] * inv;
}

// ---------------------------------------------------------------------------
extern "C" void kernel_launch(void* const* d_in, const int* in_sizes, int n_in,
                              void* d_out, int out_size, void* d_ws, size_t ws_size,
                              hipStream_t stream) {
    (void)in_sizes; (void)n_in; (void)out_size; (void)ws_size;
    const int B = 2, N = 2048, C = 768, H = 12, HD = 64;
    const float SCALE = 0.125f;        // 64^-0.5
    const float NEG = -1e30f;
    const int M   = B * N;             // 4096
    const int C3  = 3 * C;             // 2304

    const float* x      = (const float*)d_in[0];
    const int*   mask   = (const int*)  d_in[1];
    const float* w_qkv  = (const float*)d_in[2];
    const float* q_bias = (const float*)d_in[3];
    const float* v_bias = (const float*)d_in[4];
    const float* w_proj = (const float*)d_in[5];
    const float* b_proj = (const float*)d_in[6];

    float* out_main = (float*)d_out;                        // [B,N,C]
    float* out_attn = out_main + (long long)M * C;          // [B,H,N,N]

    float* ws       = (float*)d_ws;
    float* qkv_bias = ws;                                   // 2304 (pad to 4096)
    float* qkv      = ws + 4096;                            // [B*N, 3C]
    float* y        = qkv + (long long)M * C3;              // [B,N,C] head outputs

    // 1) qkv bias vector
    build_qkv_bias<<<(C3 + 255) / 256, 256, 0, stream>>>(q_bias, v_bias, qkv_bias, C);

    // 2) qkv = x @ w_qkv^T + qkv_bias        [4096 x 2304], K=768
    gemm_wmma_f32<1><<<dim3(C3 / TN, M / TM, 1), 256, 0, stream>>>(
        x, 0, 0, C,
        w_qkv, 0, 0, C,
        qkv, 0, 0, C3,
        qkv_bias, nullptr, 0,
        1.0f, NEG, M, C3, C, 1);

    // 3) scores: S[b,h] = Q K^T * scale, masked  -> attn output [24 x 2048 x 2048]
    gemm_wmma_f32<1><<<dim3(N / TN, N / TM, B * H), 256, 0, stream>>>(
        qkv,       (long long)N * C3, HD, C3,                 // Q
        qkv + C,   (long long)N * C3, HD, C3,                 // K (stored [N,hd])
        out_attn,  (long long)H * N * N, (long long)N * N, N,
        nullptr, mask, N,
        SCALE, NEG, N, N, HD, H);

    // 4) softmax rows (B*H*N rows of length N), in place
    softmax_rows<<<B * H * N, 256, 0, stream>>>(out_attn, N);

    // 5) y[b,h] = attn @ V    (K = 2048, Ncols = 64)
    gemm_wmma_f32<0><<<dim3(1, N / TM, B * H), 256, 0, stream>>>(
        out_attn,  (long long)H * N * N, (long long)N * N, N,
        qkv + 2*C, (long long)N * C3, HD, C3,                 // V as [K, hd]
        y,         (long long)N * C, HD, C,
        nullptr, nullptr, 0,
        1.0f, NEG, N, HD, N, H);

    // 6) out = y @ w_proj^T + b_proj          [4096 x 768], K=768
    gemm_wmma_f32<1><<<dim3(C / TN, M / TM, 1), 256, 0, stream>>>(
        y, 0, 0, C,
        w_proj, 0, 0, C,
        out_main, 0, 0, C,
        b_proj, nullptr, 0,
        1.0f, NEG, M, C, C, 1);
}